// PrimalCosAttention_10582799417415
// MI455X (gfx1250) — compile-verified
//
#include <hip/hip_runtime.h>
#include <hip/hip_bf16.h>
#include <math.h>

// ---------------------------------------------------------------------------
// PrimalCosAttention forward for MI455X (gfx1250, wave32).
// Memory-bound problem (~0.3 GB traffic, ~3.2 GFLOP) -> keep fp32 end-to-end
// and use V_WMMA_F32_16X16X4_F32 for all three GEMMs.
// ---------------------------------------------------------------------------

typedef float v2f __attribute__((ext_vector_type(2)));
typedef float v8f __attribute__((ext_vector_type(8)));

#define B_ 4
#define H_ 12
#define N_ 4096
#define D_ 64
#define E_ 32
#define TWOE_ 64

constexpr size_t S_ATT = (size_t)B_ * H_ * N_ * D_;   // 12582912
constexpr size_t S_SC  = (size_t)B_ * H_ * N_ * E_;   // 6291456
constexpr size_t S_WE  = (size_t)H_ * D_ * E_;        // 24576
constexpr size_t S_LAM = (size_t)H_ * E_;             // 384

constexpr size_t OFF_ATT = 0;
constexpr size_t OFF_ESC = OFF_ATT + S_ATT;
constexpr size_t OFF_RSC = OFF_ESC + S_SC;
constexpr size_t OFF_WE  = OFF_RSC + S_SC;
constexpr size_t OFF_WR  = OFF_WE  + S_WE;
constexpr size_t OFF_Q   = OFF_WR  + S_WE;
constexpr size_t OFF_K   = OFF_Q   + S_ATT;
constexpr size_t OFF_LAM = OFF_K   + S_ATT;

#define LDS_PITCH 68   // 64 + 4 pad: column reads hit distinct banks

__global__ __launch_bounds__(128) void cosattn_main(
    const float* __restrict__ Q, const float* __restrict__ K,
    const float* __restrict__ we, const float* __restrict__ wr,
    const float* __restrict__ mask, const float* __restrict__ W,
    const float* __restrict__ bias, float* __restrict__ out)
{
    __shared__ float s_xn[4][16][LDS_PITCH];  // normalized input tile, per wave
    __shared__ float s_sc[4][16][LDS_PITCH];  // [escore | rscore] tile, per wave

    const int lane = threadIdx.x & 31;
    const int wave = threadIdx.x >> 5;

    const int tiles_per_bh = N_ / 64;           // 64 blocks of 64 rows per (b,h)
    const int bh   = blockIdx.x / tiles_per_bh;
    const int tile = blockIdx.x % tiles_per_bh;
    const int b    = bh / H_;
    const int h    = bh % H_;
    const int m0   = tile * 64 + wave * 16;     // first row of this wave's tile

    float (*xn)[LDS_PITCH] = s_xn[wave];
    float (*sc)[LDS_PITCH] = s_sc[wave];

    const size_t rowbase = (size_t)bh * N_ + m0;   // flat row index in [B*H*N]
    const float* Qbh = Q + rowbase * D_;
    const float* Kbh = K + rowbase * D_;
    const float* weh = we + (size_t)h * D_ * E_;   // we[0][h] : [64][32]
    const float* wrh = wr + (size_t)h * D_ * E_;   // wr[0][h] : [64][32]

    // WMMA f32 16x16x4 lane mapping (ISA 7.12.2):
    //   A: lane l holds row m=l%16, K pair base kb = (l/16)*2 in VGPR {0,1}
    //   B: lane l holds col n=l%16, same K pair base
    //   C/D: VGPR i -> row i (+8 for lanes 16..31), col = l%16
    const int m    = lane & 15;
    const int kb   = (lane >> 4) << 1;   // 0 or 2
    const int radd = (lane >> 4) << 3;   // 0 or 8

    // ---------- phase 0: queries/escore, phase 1: keys/rscore ----------
    for (int p = 0; p < 2; ++p) {
        const float* Xbh   = p ? Kbh : Qbh;
        const float* Wh    = p ? wrh : weh;
        float*       normg = out + (p ? OFF_K   : OFF_Q)   + rowbase * D_;
        float*       scg   = out + (p ? OFF_RSC : OFF_ESC) + rowbase * E_;
        const int    scoff = p ? E_ : 0;

        // -- load 16x64 tile, L2-normalize rows (2 lanes per row) --
        {
            const int row = lane >> 1;
            const int c0  = (lane & 1) * 32;
            const float* src = Xbh + (size_t)row * D_ + c0;
            float v[32];
            float ss = 0.0f;
            #pragma unroll
            for (int i = 0; i < 32; i += 4) {
                const float4 t = *(const float4*)(src + i);
                v[i] = t.x; v[i + 1] = t.y; v[i + 2] = t.z; v[i + 3] = t.w;
                ss += t.x * t.x + t.y * t.y + t.z * t.z + t.w * t.w;
            }
            ss += __shfl_xor(ss, 1, 32);                       // pair-reduce
            const float scale = 1.0f / fmaxf(sqrtf(ss), 1e-12f); // F.normalize eps
            float* gn = normg + (size_t)row * D_ + c0;
            #pragma unroll
            for (int i = 0; i < 32; i += 4) {
                float4 t;
                t.x = v[i] * scale; t.y = v[i + 1] * scale;
                t.z = v[i + 2] * scale; t.w = v[i + 3] * scale;
                xn[row][c0 + i]     = t.x;
                xn[row][c0 + i + 1] = t.y;
                xn[row][c0 + i + 2] = t.z;
                xn[row][c0 + i + 3] = t.w;
                *(float4*)(gn + i) = t;                        // queries/keys out
            }
            asm volatile("s_wait_dscnt 0" ::: "memory");       // same-wave LDS RAW
        }

        // -- GEMM1: xn(16x64) @ Wh(64x32) via f32 WMMA, 2 N-tiles --
        v8f c0 = {};
        v8f c1 = {};
        #pragma unroll
        for (int k = 0; k < D_; k += 4) {
            v2f a;
            a.x = xn[m][k + kb];
            a.y = xn[m][k + kb + 1];
            const float* w0 = Wh + (size_t)(k + kb) * E_;      // row k+kb of [64][32]
            v2f b0, b1;
            b0.x = w0[m];          b0.y = w0[E_ + m];          // N-tile 0
            b1.x = w0[m + 16];     b1.y = w0[E_ + m + 16];     // N-tile 1
            c0 = __builtin_amdgcn_wmma_f32_16x16x4_f32(false, a, false, b0,
                                                       (short)0, c0, false, false);
            c1 = __builtin_amdgcn_wmma_f32_16x16x4_f32(false, a, false, b1,
                                                       (short)0, c1, false, false);
        }

        // -- scatter score tile: LDS concat buffer + global escore/rscore --
        #pragma unroll
        for (int i = 0; i < 8; ++i) {
            const int row = i + radd;
            sc[row][scoff + m]      = c0[i];
            sc[row][scoff + m + 16] = c1[i];
            float* g = scg + (size_t)row * E_;
            g[m]      = c0[i];
            g[m + 16] = c1[i];
        }
        asm volatile("s_wait_dscnt 0" ::: "memory");
    }

    // ---------- GEMM2: score(16x64) @ W^T(64x64), 4 N-tiles ----------
    v8f d0 = {}, d1 = {}, d2 = {}, d3 = {};
    #pragma unroll
    for (int k = 0; k < TWOE_; k += 4) {
        v2f a;
        a.x = sc[m][k + kb];
        a.y = sc[m][k + kb + 1];
        // B[k'][n] = W[n][k']  (W is [D][2E] row-major)
        const float* wt = W + (size_t)m * TWOE_ + (k + kb);
        v2f bb;
        bb.x = wt[0];              bb.y = wt[1];
        d0 = __builtin_amdgcn_wmma_f32_16x16x4_f32(false, a, false, bb,
                                                   (short)0, d0, false, false);
        bb.x = wt[16 * TWOE_];     bb.y = wt[16 * TWOE_ + 1];
        d1 = __builtin_amdgcn_wmma_f32_16x16x4_f32(false, a, false, bb,
                                                   (short)0, d1, false, false);
        bb.x = wt[32 * TWOE_];     bb.y = wt[32 * TWOE_ + 1];
        d2 = __builtin_amdgcn_wmma_f32_16x16x4_f32(false, a, false, bb,
                                                   (short)0, d2, false, false);
        bb.x = wt[48 * TWOE_];     bb.y = wt[48 * TWOE_ + 1];
        d3 = __builtin_amdgcn_wmma_f32_16x16x4_f32(false, a, false, bb,
                                                   (short)0, d3, false, false);
    }

    // ---------- epilogue: +bias, *mask, store attn ----------
    #pragma unroll
    for (int i = 0; i < 8; ++i) {
        const int row = i + radd;
        const float mv = mask[(size_t)b * N_ + m0 + row];
        float* g = out + OFF_ATT + (rowbase + row) * D_;
        g[m]      = (d0[i] + bias[m])      * mv;
        g[m + 16] = (d1[i] + bias[m + 16]) * mv;
        g[m + 32] = (d2[i] + bias[m + 32]) * mv;
        g[m + 48] = (d3[i] + bias[m + 48]) * mv;
    }
}

// Tiny pass-through outputs: we[0], wr[0], Lambda.
__global__ void cosattn_tail(const float* __restrict__ we,
                             const float* __restrict__ wr,
                             const float* __restrict__ lam,
                             float* __restrict__ out)
{
    const size_t i = (size_t)blockIdx.x * blockDim.x + threadIdx.x;
    if (i < S_WE) {
        out[OFF_WE + i] = we[i];   // first H*D*E elements of [2,H,D,E] == we[0]
        out[OFF_WR + i] = wr[i];
    }
    if (i < S_LAM) out[OFF_LAM + i] = lam[i];
}

extern "C" void kernel_launch(void* const* d_in, const int* in_sizes, int n_in,
                              void* d_out, int out_size, void* d_ws, size_t ws_size,
                              hipStream_t stream) {
    const float* Q    = (const float*)d_in[0];
    const float* K    = (const float*)d_in[1];
    const float* we   = (const float*)d_in[2];
    const float* wr   = (const float*)d_in[3];
    const float* mask = (const float*)d_in[4];
    const float* W    = (const float*)d_in[5];
    const float* bias = (const float*)d_in[6];
    const float* lam  = (const float*)d_in[7];
    float* out = (float*)d_out;

    const unsigned grid = (unsigned)(B_ * H_ * (N_ / 64));   // 3072 blocks, 4 waves each
    cosattn_main<<<grid, 128, 0, stream>>>(Q, K, we, wr, mask, W, bias, out);

    const unsigned tgrid = (unsigned)((S_WE + 255) / 256);
    cosattn_tail<<<tgrid, 256, 0, stream>>>(we, wr, lam, out);
}